// EnhancedGNN_87376814670438
// MI455X (gfx1250) — compile-verified
//
#include <hip/hip_runtime.h>
#include <math.h>

#define NN   50000
#define EE   600000
#define DIN  64
#define HH   128
#define EDD  9
#define OUTD 2

typedef float v2f __attribute__((ext_vector_type(2)));
typedef float v8f __attribute__((ext_vector_type(8)));

// ---------------------------------------------------------------------------
// C[M x 128] = A[M x K] * W[K x 128]  (row-major, fp32, WMMA 16x16x4)
// One block = 16 rows x 128 cols; 8 waves, each owns a 16x16 tile.
// ---------------------------------------------------------------------------
template <int K>
__global__ __launch_bounds__(256) void wmma_gemm_f32(const float* __restrict__ A,
                                                     const float* __restrict__ W,
                                                     float* __restrict__ C) {
    constexpr int STRIDE = K + 4;               // LDS pad to spread banks
    __shared__ float As[16 * STRIDE];

    const int rowBase = blockIdx.x * 16;
    // cooperative load of the 16 x K A tile
    for (int i = threadIdx.x; i < 16 * K; i += 256) {
        int r = i / K, c = i - r * K;
        As[r * STRIDE + c] = A[(size_t)(rowBase + r) * K + c];
    }
    __syncthreads();

    const int wave  = threadIdx.x >> 5;
    const int lane  = threadIdx.x & 31;
    const int nloc  = lane & 15;                // N index inside tile
    const int col   = wave * 16 + nloc;         // global output column
    const int khalf = (lane >> 4) << 1;         // lanes 0-15 -> K+0/1, 16-31 -> K+2/3

    v8f c;
#pragma unroll
    for (int i = 0; i < 8; ++i) c[i] = 0.0f;

#pragma unroll
    for (int k = 0; k < K; k += 4) {
        v2f a, b;
        const float* ap = &As[nloc * STRIDE + k + khalf];
        a[0] = ap[0];
        a[1] = ap[1];
        b[0] = W[(size_t)(k + khalf) * HH + col];
        b[1] = W[(size_t)(k + khalf + 1) * HH + col];
        c = __builtin_amdgcn_wmma_f32_16x16x4_f32(false, a, false, b,
                                                  (short)0, c, false, false);
    }

    const int rowOff = (lane >> 4) << 3;        // lanes 16-31 hold rows +8
#pragma unroll
    for (int v = 0; v < 8; ++v)
        C[(size_t)(rowBase + v + rowOff) * HH + col] = c[v];
}

// ---------------------------------------------------------------------------
// Per-node: a_src/a_dst dot products (wave per node) + init amax/denom/deg
// ---------------------------------------------------------------------------
__global__ __launch_bounds__(256) void node_avec_init(const float* __restrict__ xl,
                                                      const float* __restrict__ att_src,
                                                      const float* __restrict__ att_dst,
                                                      float* __restrict__ a_src,
                                                      float* __restrict__ a_dst,
                                                      float* __restrict__ amax,
                                                      float* __restrict__ denom,
                                                      float* __restrict__ deg) {
    int gid  = blockIdx.x * 256 + threadIdx.x;
    int n    = gid >> 5;
    int lane = threadIdx.x & 31;
    if (n >= NN) return;
    float4 xv  = ((const float4*)(xl + (size_t)n * HH))[lane];
    float4 asv = ((const float4*)att_src)[lane];
    float4 adv = ((const float4*)att_dst)[lane];
    float s = xv.x * asv.x + xv.y * asv.y + xv.z * asv.z + xv.w * asv.w;
    float d = xv.x * adv.x + xv.y * adv.y + xv.z * adv.z + xv.w * adv.w;
    for (int off = 16; off > 0; off >>= 1) {
        s += __shfl_down(s, off);
        d += __shfl_down(d, off);
    }
    if (lane == 0) {
        a_src[n] = s;
        a_dst[n] = d;
        amax[n]  = -INFINITY;
        denom[n] = 0.0f;
        deg[n]   = 0.0f;
    }
}

// h[n*128 + c] = bias[c]
__global__ void init_bias_mat(float* __restrict__ h, const float* __restrict__ bias) {
    int i = blockIdx.x * 256 + threadIdx.x;
    if (i < NN * HH) h[i] = bias[i & (HH - 1)];
}

// w_e[j] = sum_k W_edge[j][k] * att_edge[k]
__global__ void prep_we(const float* __restrict__ W_edge,
                        const float* __restrict__ att_edge,
                        float* __restrict__ w_e) {
    int j = threadIdx.x;
    if (j >= EDD) return;
    float acc = 0.0f;
    for (int k = 0; k < HH; ++k) acc += W_edge[j * HH + k] * att_edge[k];
    w_e[j] = acc;
}

// alpha = leaky_relu(a_src[src] + a_dst[dst] + edge_attr . w_e); atomicMax amax[dst]
__global__ void edge_alpha(const int* __restrict__ ei,
                           const float* __restrict__ edge_attr,
                           const float* __restrict__ w_e,
                           const float* __restrict__ a_src,
                           const float* __restrict__ a_dst,
                           float* __restrict__ alpha_buf,
                           float* __restrict__ amax) {
    int e = blockIdx.x * 256 + threadIdx.x;
    if (e >= EE) return;
    int s = ei[e], d = ei[EE + e];
    float ae = 0.0f;
#pragma unroll
    for (int j = 0; j < EDD; ++j) ae += edge_attr[(size_t)e * EDD + j] * w_e[j];
    float alpha = a_src[s] + a_dst[d] + ae;
    alpha = alpha > 0.0f ? alpha : 0.2f * alpha;
    alpha_buf[e] = alpha;
    atomicMax(&amax[d], alpha);
}

// ex = exp(alpha - amax[dst]); denom[dst] += ex; deg[dst] += 1
__global__ void edge_exp(const int* __restrict__ ei,
                         float* __restrict__ ex_buf,
                         const float* __restrict__ amax,
                         float* __restrict__ denom,
                         float* __restrict__ deg) {
    int e = blockIdx.x * 256 + threadIdx.x;
    if (e >= EE) return;
    int d = ei[EE + e];
    float ex = __expf(ex_buf[e] - amax[d]);
    ex_buf[e] = ex;
    atomicAdd(&denom[d], ex);
    atomicAdd(&deg[d], 1.0f);
}

// att = ex/denom[dst]; h1[dst] += att * xl[src]   (one wave per edge)
__global__ __launch_bounds__(256) void edge_att_scatter(const int* __restrict__ ei,
                                                        const float* __restrict__ ex_buf,
                                                        const float* __restrict__ denom,
                                                        const float* __restrict__ xl,
                                                        float* __restrict__ h1,
                                                        float* __restrict__ att_out) {
    int gid  = blockIdx.x * 256 + threadIdx.x;
    int e    = gid >> 5;
    int lane = threadIdx.x & 31;
    if (e >= EE) return;
    int s = ei[e], d = ei[EE + e];
    float att = ex_buf[e] / denom[d];
    if (lane == 0) att_out[e] = att;
    float4 xv = ((const float4*)(xl + (size_t)s * HH))[lane];
    float* hp = h1 + (size_t)d * HH + lane * 4;
    atomicAdd(hp + 0, xv.x * att);
    atomicAdd(hp + 1, xv.y * att);
    atomicAdd(hp + 2, xv.z * att);
    atomicAdd(hp + 3, xv.w * att);
}

// dis[n] = deg>0 ? rsqrt(max(deg,1)) : 0
__global__ void node_dis(const float* __restrict__ deg, float* __restrict__ dis) {
    int n = blockIdx.x * 256 + threadIdx.x;
    if (n < NN) {
        float dg = deg[n];
        dis[n] = dg > 0.0f ? rsqrtf(fmaxf(dg, 1.0f)) : 0.0f;
    }
}

// h2[dst] += dis[src]*dis[dst] * x2[src]  (one wave per edge)
__global__ __launch_bounds__(256) void gcn_scatter(const int* __restrict__ ei,
                                                   const float* __restrict__ dis,
                                                   const float* __restrict__ x2,
                                                   float* __restrict__ h2) {
    int gid  = blockIdx.x * 256 + threadIdx.x;
    int e    = gid >> 5;
    int lane = threadIdx.x & 31;
    if (e >= EE) return;
    int s = ei[e], d = ei[EE + e];
    float norm = dis[s] * dis[d];
    float4 xv = ((const float4*)(x2 + (size_t)s * HH))[lane];
    float* hp = h2 + (size_t)d * HH + lane * 4;
    atomicAdd(hp + 0, xv.x * norm);
    atomicAdd(hp + 1, xv.y * norm);
    atomicAdd(hp + 2, xv.z * norm);
    atomicAdd(hp + 3, xv.w * norm);
}

// out[n] = relu(h2[n]) @ W_out + b_out   (wave per node)
__global__ __launch_bounds__(256) void out_head(const float* __restrict__ h2,
                                                const float* __restrict__ W_out,
                                                const float* __restrict__ b_out,
                                                float* __restrict__ out) {
    int gid  = blockIdx.x * 256 + threadIdx.x;
    int n    = gid >> 5;
    int lane = threadIdx.x & 31;
    if (n >= NN) return;
    float4 hv = ((const float4*)(h2 + (size_t)n * HH))[lane];
    hv.x = fmaxf(hv.x, 0.0f);
    hv.y = fmaxf(hv.y, 0.0f);
    hv.z = fmaxf(hv.z, 0.0f);
    hv.w = fmaxf(hv.w, 0.0f);
    const float2* Wp = (const float2*)W_out;    // [H][2]
    float2 w0 = Wp[4 * lane + 0], w1 = Wp[4 * lane + 1];
    float2 w2 = Wp[4 * lane + 2], w3 = Wp[4 * lane + 3];
    float a0 = hv.x * w0.x + hv.y * w1.x + hv.z * w2.x + hv.w * w3.x;
    float a1 = hv.x * w0.y + hv.y * w1.y + hv.z * w2.y + hv.w * w3.y;
    for (int off = 16; off > 0; off >>= 1) {
        a0 += __shfl_down(a0, off);
        a1 += __shfl_down(a1, off);
    }
    if (lane == 0) {
        out[n * OUTD + 0] = a0 + b_out[0];
        out[n * OUTD + 1] = a1 + b_out[1];
    }
}

// edge_index (int32) -> float, into d_out section
__global__ void ei_copy(const int* __restrict__ ei, float* __restrict__ dst) {
    int i = blockIdx.x * 256 + threadIdx.x;
    if (i < 2 * EE) dst[i] = (float)ei[i];
}

// ---------------------------------------------------------------------------
extern "C" void kernel_launch(void* const* d_in, const int* in_sizes, int n_in,
                              void* d_out, int out_size, void* d_ws, size_t ws_size,
                              hipStream_t stream) {
    const float* x        = (const float*)d_in[0];
    const int*   ei       = (const int*)d_in[1];
    const float* edge_attr= (const float*)d_in[2];
    const float* W_gat    = (const float*)d_in[3];
    const float* att_src  = (const float*)d_in[4];
    const float* att_dst  = (const float*)d_in[5];
    const float* W_edge   = (const float*)d_in[6];
    const float* att_edge = (const float*)d_in[7];
    const float* bias_gat = (const float*)d_in[8];
    const float* W_gcn    = (const float*)d_in[9];
    const float* bias_gcn = (const float*)d_in[10];
    const float* W_out    = (const float*)d_in[11];
    const float* b_out    = (const float*)d_in[12];

    // workspace layout (floats)
    float* ws    = (float*)d_ws;
    float* bufA  = ws;                           // xl, later x2  (N*H)
    float* bufB  = bufA + (size_t)NN * HH;       // h1, later h2  (N*H)
    float* exb   = bufB + (size_t)NN * HH;       // alpha -> ex   (E)
    float* a_src = exb + EE;                     // N
    float* a_dst = a_src + NN;                   // N
    float* amax  = a_dst + NN;                   // N
    float* denom = amax + NN;                    // N
    float* deg   = denom + NN;                   // N
    float* dis   = deg + NN;                     // N
    float* w_e   = dis + NN;                     // 16

    // output layout: out [N*2] | edge_index [2E] | att [E]
    float* out_p = (float*)d_out;
    float* ei_p  = out_p + NN * OUTD;
    float* att_p = ei_p + 2 * EE;

    const int nodeWaveBlocks = (NN * 32 + 255) / 256;   // wave-per-node kernels
    const int edgeWaveBlocks = (EE * 32 + 255) / 256;   // wave-per-edge kernels
    const int edgeBlocks     = (EE + 255) / 256;
    const int matBlocks      = (NN * HH + 255) / 256;

    // --- GAT ---
    wmma_gemm_f32<DIN><<<NN / 16, 256, 0, stream>>>(x, W_gat, bufA);          // xl
    prep_we<<<1, 16, 0, stream>>>(W_edge, att_edge, w_e);
    node_avec_init<<<nodeWaveBlocks, 256, 0, stream>>>(bufA, att_src, att_dst,
                                                       a_src, a_dst, amax, denom, deg);
    init_bias_mat<<<matBlocks, 256, 0, stream>>>(bufB, bias_gat);             // h1 = bias
    edge_alpha<<<edgeBlocks, 256, 0, stream>>>(ei, edge_attr, w_e, a_src, a_dst, exb, amax);
    edge_exp<<<edgeBlocks, 256, 0, stream>>>(ei, exb, amax, denom, deg);
    edge_att_scatter<<<edgeWaveBlocks, 256, 0, stream>>>(ei, exb, denom, bufA, bufB, att_p);

    // --- GCN ---
    wmma_gemm_f32<HH><<<NN / 16, 256, 0, stream>>>(bufB, W_gcn, bufA);        // x2 = h1 @ W_gcn
    node_dis<<<(NN + 255) / 256, 256, 0, stream>>>(deg, dis);
    init_bias_mat<<<matBlocks, 256, 0, stream>>>(bufB, bias_gcn);             // h2 = bias
    gcn_scatter<<<edgeWaveBlocks, 256, 0, stream>>>(ei, dis, bufA, bufB);

    // --- head + aux outputs ---
    out_head<<<nodeWaveBlocks, 256, 0, stream>>>(bufB, W_out, b_out, out_p);
    ei_copy<<<(2 * EE + 255) / 256, 256, 0, stream>>>(ei, ei_p);
}